// STAGNN_82197084110973
// MI455X (gfx1250) — compile-verified
//
#include <hip/hip_runtime.h>

#define NF   128
#define HID  64
#define NC   32
#define HC   2048   // HID*NC
#define KHOPS 4
#define CSTF 1e-5f

typedef __attribute__((ext_vector_type(2))) float v2f;
typedef __attribute__((ext_vector_type(8))) float v8f;

// V_WMMA_F32_16X16X4_F32 : D[16x16] = A[16x4] * B[4x16] + C
__device__ __forceinline__ v8f wmma4f32(v2f a, v2f b, v8f c) {
    return __builtin_amdgcn_wmma_f32_16x16x4_f32(
        /*neg_a=*/false, a, /*neg_b=*/false, b,
        /*c_mod=*/(short)0, c, /*reuse_a=*/false, /*reuse_b=*/false);
}

// ---------------------------------------------------------------- zero (also used for int buffers)
__global__ void k_zero4(float4* __restrict__ p, long long n4) {
    long long i = (long long)blockIdx.x * blockDim.x + threadIdx.x;
    if (i < n4) p[i] = make_float4(0.f, 0.f, 0.f, 0.f);
}

// ---------------------------------------------------------------- CSR build: count, deg_inv, scan, fill
__global__ void k_counti(const int* __restrict__ col, int* __restrict__ cnt, int E) {
    int e = blockIdx.x * blockDim.x + threadIdx.x;
    if (e < E) atomicAdd(&cnt[col[e]], 1);
}
__global__ void k_deginv(const int* __restrict__ cnt, float* __restrict__ dinv, int n) {
    int i = blockIdx.x * blockDim.x + threadIdx.x;
    if (i < n) { int d = cnt[i]; dinv[i] = (d > 0) ? (1.0f / (float)d) : 0.0f; }
}
// block-wise exclusive scan (Hillis-Steele in LDS), 256/block
__global__ __launch_bounds__(256) void k_scan1(const int* __restrict__ cnt,
                                               int* __restrict__ excl,
                                               int* __restrict__ bsum, int n) {
    __shared__ int s[256];
    int t = threadIdx.x;
    int i = blockIdx.x * 256 + t;
    int v = (i < n) ? cnt[i] : 0;
    s[t] = v;
    __syncthreads();
    for (int off = 1; off < 256; off <<= 1) {
        int y = (t >= off) ? s[t - off] : 0;
        int x = s[t];
        __syncthreads();
        s[t] = x + y;
        __syncthreads();
    }
    if (i < n) excl[i] = s[t] - v;
    if (t == 255) bsum[blockIdx.x] = s[255];
}
__global__ __launch_bounds__(256) void k_scan2(int* __restrict__ bsum, int nb) {
    __shared__ int s[256];
    int t = threadIdx.x;
    int v = (t < nb) ? bsum[t] : 0;
    s[t] = v;
    __syncthreads();
    for (int off = 1; off < 256; off <<= 1) {
        int y = (t >= off) ? s[t - off] : 0;
        int x = s[t];
        __syncthreads();
        s[t] = x + y;
        __syncthreads();
    }
    if (t < nb) bsum[t] = s[t] - v;
}
__global__ void k_scan3(int* __restrict__ excl, const int* __restrict__ bsum, int n) {
    int i = blockIdx.x * blockDim.x + threadIdx.x;
    if (i < n) excl[i] += bsum[i >> 8];
}
__global__ void k_fill(const int* __restrict__ row, const int* __restrict__ col,
                       const int* __restrict__ rowptr, int* __restrict__ cursor,
                       const float* __restrict__ dinv,
                       int* __restrict__ srcidx, float* __restrict__ wnorm, int E) {
    int e = blockIdx.x * blockDim.x + threadIdx.x;
    if (e >= E) return;
    int c = col[e], r = row[e];
    int pos = rowptr[c] + atomicAdd(&cursor[c], 1);
    srcidx[pos] = r;
    wnorm[pos]  = dinv[r];
}

// ---------------------------------------------------------------- fused node MLP (WMMA f32)
__global__ __launch_bounds__(32) void k_mlp(
    const float* __restrict__ feat,
    const float* __restrict__ Win, const float* __restrict__ bin,
    const float* __restrict__ WQm, const float* __restrict__ bQ,
    const float* __restrict__ WKm, const float* __restrict__ bK,
    const float* __restrict__ WVm, const float* __restrict__ bV,
    const float* __restrict__ hopwise,
    float* __restrict__ Q, float* __restrict__ Kf, float* __restrict__ V,
    float* __restrict__ out, int ntiles)
{
    __shared__ float xs[16 * HID];
    int tile = blockIdx.x;
    if (tile >= ntiles) return;
    int lane  = threadIdx.x & 31;
    int mrow  = lane & 15;
    int khalf = (lane >> 4) << 1;   // 0 or 2
    int rhalf = (lane >> 4) << 3;   // 0 or 8
    long long base = (long long)tile * 16;

    // GEMM1: x = relu(feat @ Win + bin)
    v8f acc[4] = {};
    const float* frow = feat + base * NF;
    for (int k = 0; k < NF; k += 4) {
        v2f a;
        a.x = frow[mrow * NF + k + khalf];
        a.y = frow[mrow * NF + k + khalf + 1];
#pragma unroll
        for (int t = 0; t < 4; ++t) {
            v2f b;
            b.x = Win[(k + khalf)     * HID + t * 16 + mrow];
            b.y = Win[(k + khalf + 1) * HID + t * 16 + mrow];
            acc[t] = wmma4f32(a, b, acc[t]);
        }
    }
#pragma unroll
    for (int t = 0; t < 4; ++t) {
        int cc = t * 16 + mrow;
        float bb = bin[cc];
#pragma unroll
        for (int r = 0; r < 8; ++r) {
            float xv = acc[t][r] + bb;
            xs[(rhalf + r) * HID + cc] = xv > 0.f ? xv : 0.f;
        }
    }
    __syncthreads();

    // GEMM2: Q = 1 + elu(x @ WQ + bQ)
    v8f qa[4] = {};
    for (int k = 0; k < HID; k += 4) {
        v2f a;
        a.x = xs[mrow * HID + k + khalf];
        a.y = xs[mrow * HID + k + khalf + 1];
#pragma unroll
        for (int t = 0; t < 4; ++t) {
            v2f b;
            b.x = WQm[(k + khalf)     * HID + t * 16 + mrow];
            b.y = WQm[(k + khalf + 1) * HID + t * 16 + mrow];
            qa[t] = wmma4f32(a, b, qa[t]);
        }
    }
#pragma unroll
    for (int t = 0; t < 4; ++t) {
        int cc = t * 16 + mrow;
        float bb = bQ[cc];
#pragma unroll
        for (int r = 0; r < 8; ++r) {
            float z = qa[t][r] + bb;
            Q[(base + rhalf + r) * HID + cc] = (z > 0.f) ? (z + 1.0f) : __expf(z);
        }
    }

    // GEMM3: Kf = 1 + elu(x @ WK + bK)
    v8f ka[4] = {};
    for (int k = 0; k < HID; k += 4) {
        v2f a;
        a.x = xs[mrow * HID + k + khalf];
        a.y = xs[mrow * HID + k + khalf + 1];
#pragma unroll
        for (int t = 0; t < 4; ++t) {
            v2f b;
            b.x = WKm[(k + khalf)     * HID + t * 16 + mrow];
            b.y = WKm[(k + khalf + 1) * HID + t * 16 + mrow];
            ka[t] = wmma4f32(a, b, ka[t]);
        }
    }
#pragma unroll
    for (int t = 0; t < 4; ++t) {
        int cc = t * 16 + mrow;
        float bb = bK[cc];
#pragma unroll
        for (int r = 0; r < 8; ++r) {
            float z = ka[t][r] + bb;
            Kf[(base + rhalf + r) * HID + cc] = (z > 0.f) ? (z + 1.0f) : __expf(z);
        }
    }

    // GEMM4: V = x @ WV + bV ; hidden init = hopwise[0]*V
    v8f va[2] = {};
    for (int k = 0; k < HID; k += 4) {
        v2f a;
        a.x = xs[mrow * HID + k + khalf];
        a.y = xs[mrow * HID + k + khalf + 1];
#pragma unroll
        for (int t = 0; t < 2; ++t) {
            v2f b;
            b.x = WVm[(k + khalf)     * NC + t * 16 + mrow];
            b.y = WVm[(k + khalf + 1) * NC + t * 16 + mrow];
            va[t] = wmma4f32(a, b, va[t]);
        }
    }
    float hw0 = hopwise[0];
#pragma unroll
    for (int t = 0; t < 2; ++t) {
        int cc = t * 16 + mrow;
        float bb = bV[cc];
#pragma unroll
        for (int r = 0; r < 8; ++r) {
            float vv = va[t][r] + bb;
            long long node = base + rhalf + r;
            V[node * NC + cc]   = vv;
            out[node * NC + cc] = hw0 * vv;
        }
    }
}

// ---------------------------------------------------------------- M0[i] = Kf[i] (x) V[i] ; tK += Kf[i]
__global__ __launch_bounds__(256) void k_outer(
    const float* __restrict__ Kf, const float* __restrict__ V,
    float* __restrict__ M0, float* __restrict__ tK, int n)
{
    int i = blockIdx.x;
    if (i >= n) return;
    int t = threadIdx.x;
    const float* kf = Kf + (long long)i * HID;
    const float* vv = V  + (long long)i * NC;
    float*       m  = M0 + (long long)i * HC;
#pragma unroll
    for (int j = 0; j < 8; ++j) {
        int idx = t + 256 * j;
        int h = idx >> 5, c = idx & 31;
        m[idx] = kf[h] * vv[c];
    }
    if (t < HID) atomicAdd(&tK[t], kf[t]);
}

// ---------------------------------------------------------------- tM = Kf^T @ V via WMMA split-K
__global__ __launch_bounds__(32) void k_tmat(
    const float* __restrict__ Kf, const float* __restrict__ V,
    float* __restrict__ tM, int n, int chunk)
{
    int lane  = threadIdx.x & 31;
    int mrow  = lane & 15;
    int khalf = (lane >> 4) << 1;
    int rhalf = (lane >> 4) << 3;
    long long k0 = (long long)blockIdx.x * chunk;
    if (k0 >= n) return;

    v8f acc[4][2] = {};
    for (int kk = 0; kk < chunk; kk += 4) {
        long long nb = k0 + kk;
        if (nb + 4 > n) break;
        v2f b0, b1;
        b0.x = V[(nb + khalf)     * NC + mrow];
        b0.y = V[(nb + khalf + 1) * NC + mrow];
        b1.x = V[(nb + khalf)     * NC + 16 + mrow];
        b1.y = V[(nb + khalf + 1) * NC + 16 + mrow];
#pragma unroll
        for (int mt = 0; mt < 4; ++mt) {
            v2f a;   // A[h][node] = Kf^T
            a.x = Kf[(nb + khalf)     * HID + mt * 16 + mrow];
            a.y = Kf[(nb + khalf + 1) * HID + mt * 16 + mrow];
            acc[mt][0] = wmma4f32(a, b0, acc[mt][0]);
            acc[mt][1] = wmma4f32(a, b1, acc[mt][1]);
        }
    }
#pragma unroll
    for (int mt = 0; mt < 4; ++mt)
#pragma unroll
        for (int ct = 0; ct < 2; ++ct)
#pragma unroll
            for (int r = 0; r < 8; ++r) {
                int h = mt * 16 + rhalf + r;
                int c = ct * 16 + mrow;
                atomicAdd(&tM[h * NC + c], acc[mt][ct][r]);
            }
}

// ---------------------------------------------------------------- fused hop: CSR gather + combine
// one block per destination node; M state held in registers (256 thr x 8 f32)
// float4 mapping: thread t holds c = 4*(t&7)+k (k=0..3), h = t>>3 and t>>3+32
__global__ __launch_bounds__(256) void k_hop(
    const int* __restrict__ rowptr, const int* __restrict__ cnt,
    const int* __restrict__ srcidx, const float* __restrict__ wnorm,
    const float* __restrict__ Mo, const float* __restrict__ Ko,
    const float* __restrict__ Q, const float* __restrict__ hopwise, int hop,
    float* __restrict__ Mn, float* __restrict__ Kn,
    float* __restrict__ out, int n)
{
    __shared__ int   s_src[128];
    __shared__ float s_w[128];
    __shared__ float s_q[HID];
    __shared__ float s_kf[HID];
    __shared__ float s_red[256 * 4];

    int i = blockIdx.x;
    if (i >= n) return;
    int t = threadIdx.x;

    int start = rowptr[i];
    int len   = cnt[i];

    float4 a0 = make_float4(0.f, 0.f, 0.f, 0.f);
    float4 a1 = a0;
    float4 aK = a0;   // threads 0..15 hold Kf_new[t*4 .. t*4+3]

    for (int e0 = 0; e0 < len; e0 += 128) {
        int nch = (len - e0 < 128) ? (len - e0) : 128;
        __syncthreads();
        if (t < nch) {
            s_src[t] = srcidx[start + e0 + t];
            s_w[t]   = wnorm[start + e0 + t];
        }
        __syncthreads();
        for (int ee = 0; ee < nch; ++ee) {
            // prefetch row of edge 4 ahead (256 thr x 32B = 8KB row)
            if (ee + 4 < nch) {
                const char* pf = (const char*)(Mo + (long long)s_src[ee + 4] * HC);
                __builtin_prefetch(pf + t * 32, 0, 0);
            }
            int   src = s_src[ee];
            float w   = s_w[ee];
            const float4* mo = (const float4*)(Mo + (long long)src * HC);
            float4 v0 = mo[t];
            float4 v1 = mo[t + 256];
            a0.x += w * v0.x; a0.y += w * v0.y; a0.z += w * v0.z; a0.w += w * v0.w;
            a1.x += w * v1.x; a1.y += w * v1.y; a1.z += w * v1.z; a1.w += w * v1.w;
            if (t < 16) {
                float4 kv = ((const float4*)(Ko + (long long)src * HID))[t];
                aK.x += w * kv.x; aK.y += w * kv.y; aK.z += w * kv.z; aK.w += w * kv.w;
            }
        }
    }

    // write propagated state (plain stores, no zero-fill needed)
    float4* mn = (float4*)(Mn + (long long)i * HC);
    mn[t]       = a0;
    mn[t + 256] = a1;
    if (t < 16) ((float4*)(Kn + (long long)i * HID))[t] = aK;

    __syncthreads();
    if (t < 16) ((float4*)s_kf)[t] = aK;
    if (t < HID) s_q[t] = Q[(long long)i * HID + t];
    __syncthreads();

    // partial H contributions: this thread's two h-rows
    int h0 = t >> 3;
    float q0 = s_q[h0], q1 = s_q[h0 + 32];
    s_red[t * 4 + 0] = q0 * a0.x + q1 * a1.x;
    s_red[t * 4 + 1] = q0 * a0.y + q1 * a1.y;
    s_red[t * 4 + 2] = q0 * a0.z + q1 * a1.z;
    s_red[t * 4 + 3] = q0 * a0.w + q1 * a1.w;
    __syncthreads();

    if (t < NC) {
        // output class c = t : contributing thread m*8+(t>>2) holds element t&3
        // -> LDS addr m*32 + t (conflict-free)
        float H = 0.f;
#pragma unroll
        for (int m = 0; m < 32; ++m) H += s_red[m * 32 + t];
        float C = 0.f;
#pragma unroll
        for (int h = 0; h < HID; ++h) C += s_q[h] * s_kf[h];
        float hw = hopwise[hop + 1];
        out[(long long)i * NC + t] += hw * H / (C + CSTF);
    }
}

// ---------------------------------------------------------------- teleport finalize
__global__ __launch_bounds__(256) void k_teleport(
    const float* __restrict__ Q, const float* __restrict__ tM,
    const float* __restrict__ tK, const float* __restrict__ tele,
    float* __restrict__ out, int n)
{
    int w = threadIdx.x >> 5, lane = threadIdx.x & 31;
    int i = blockIdx.x * 8 + w;
    if (i >= n) return;
    const float* q = Q + (long long)i * HID;
    float num = 0.f, den = 0.f;
    for (int h = 0; h < HID; ++h) {
        float qh = q[h];
        num += qh * tM[h * NC + lane];
        den += qh * tK[h];
    }
    float invn = 1.0f / (float)n;
    float res = (num * invn) / (den * invn + CSTF);
    out[(long long)i * NC + lane] += tele[0] * res;
}

// ================================================================ host
extern "C" void kernel_launch(void* const* d_in, const int* in_sizes, int n_in,
                              void* d_out, int out_size, void* d_ws, size_t ws_size,
                              hipStream_t stream)
{
    const float* feat = (const float*)d_in[0];
    const float* Win  = (const float*)d_in[1];
    const float* bin  = (const float*)d_in[2];
    const float* WQm  = (const float*)d_in[3];
    const float* bQ   = (const float*)d_in[4];
    const float* WKm  = (const float*)d_in[5];
    const float* bK   = (const float*)d_in[6];
    const float* WVm  = (const float*)d_in[7];
    const float* bV   = (const float*)d_in[8];
    const float* hopw = (const float*)d_in[9];
    const float* tele = (const float*)d_in[10];
    const int*   ei   = (const int*)d_in[11];

    int n = in_sizes[0] / NF;
    int E = in_sizes[11] / 2;
    const int* row = ei;
    const int* col = ei + E;
    float* out = (float*)d_out;

    // workspace carve (floats/ints share, 256B aligned slices)
    char* ws = (char*)d_ws;
    size_t off = 0;
    auto alloc = [&](size_t bytes) { void* p = ws + off; off += (bytes + 255) & ~(size_t)255; return p; };
    int*   cnt    = (int*)  alloc((size_t)n * 4);
    int*   rowptr = (int*)  alloc((size_t)n * 4);
    int*   cursor = (int*)  alloc((size_t)n * 4);
    int*   bsum   = (int*)  alloc(256 * 4);
    int*   srcidx = (int*)  alloc((size_t)E * 4);
    float* wnorm  = (float*)alloc((size_t)E * 4);
    float* dinv   = (float*)alloc((size_t)n * 4);
    float* Qb     = (float*)alloc((size_t)n * HID * 4);
    float* KfA    = (float*)alloc((size_t)n * HID * 4);
    float* KfB    = (float*)alloc((size_t)n * HID * 4);
    float* Vb     = (float*)alloc((size_t)n * NC * 4);
    float* MA     = (float*)alloc((size_t)n * HC * 4);
    float* MB     = (float*)alloc((size_t)n * HC * 4);
    float* tM     = (float*)alloc((size_t)HC * 4);
    float* tK     = (float*)alloc((size_t)HID * 4);
    (void)ws_size; (void)n_in; (void)out_size;

    auto zero = [&](void* p, long long cnt_f32) {
        long long n4 = cnt_f32 / 4;
        unsigned g = (unsigned)((n4 + 255) / 256);
        k_zero4<<<g, 256, 0, stream>>>((float4*)p, n4);
    };

    // ---- CSR build
    zero(cnt, n);
    zero(cursor, n);
    zero(tM, HC);
    zero(tK, HID);
    k_counti<<<(E + 255) / 256, 256, 0, stream>>>(col, cnt, E);
    k_deginv<<<(n + 255) / 256, 256, 0, stream>>>(cnt, dinv, n);
    int nb = (n + 255) / 256;
    k_scan1<<<nb, 256, 0, stream>>>(cnt, rowptr, bsum, n);
    k_scan2<<<1, 256, 0, stream>>>(bsum, nb);
    k_scan3<<<nb, 256, 0, stream>>>(rowptr, bsum, n);
    k_fill<<<(E + 255) / 256, 256, 0, stream>>>(row, col, rowptr, cursor, dinv, srcidx, wnorm, E);

    // ---- node MLP (WMMA): Q, Kf, V, out = hopwise[0]*V
    int ntiles = n / 16;
    k_mlp<<<ntiles, 32, 0, stream>>>(feat, Win, bin, WQm, bQ, WKm, bK, WVm, bV,
                                     hopw, Qb, KfA, Vb, out, ntiles);

    // ---- M0 outer products + teleport sums (tM via WMMA split-K GEMM Kf^T @ V)
    k_outer<<<n, 256, 0, stream>>>(KfA, Vb, MA, tK, n);
    {
        int B = 125;
        int chunk = ((n + B - 1) / B + 3) & ~3;
        k_tmat<<<B, 32, 0, stream>>>(KfA, Vb, tM, n, chunk);
    }

    // ---- K-hop propagation (fused gather + combine, ping-pong)
    float* Mc = MA; float* Mx = MB;
    float* Kc = KfA; float* Kx = KfB;
    for (int hop = 0; hop < KHOPS; ++hop) {
        k_hop<<<n, 256, 0, stream>>>(rowptr, cnt, srcidx, wnorm, Mc, Kc,
                                     Qb, hopw, hop, Mx, Kx, out, n);
        float* tp;
        tp = Mc; Mc = Mx; Mx = tp;
        tp = Kc; Kc = Kx; Kx = tp;
    }

    // ---- teleport branch
    k_teleport<<<(n + 7) / 8, 256, 0, stream>>>(Qb, tM, tK, tele, out, n);
}